// BurgerDissipativeLossOperator_34849364639903
// MI455X (gfx1250) — compile-verified
//
#include <hip/hip_runtime.h>
#include <hip/hip_bf16.h>
#include <stdint.h>

#define NTHREADS 256
#define TILE 512            // edges per tile (2 edges per lane)
#define DELTA_T 0.01f
#define MU 0.01f

typedef float    vf2 __attribute__((ext_vector_type(2)));
typedef float    vf4 __attribute__((ext_vector_type(4)));
typedef unsigned vu2 __attribute__((ext_vector_type(2)));

// ---- gfx1250 async global->LDS copies (ASYNCcnt-tracked), per CDNA5 ISA ----
__device__ __forceinline__ void async_ld_b64(uint32_t lds_off, const void* gaddr) {
    asm volatile("global_load_async_to_lds_b64 %0, %1, off"
                 :: "v"(lds_off), "v"(gaddr) : "memory");
}
__device__ __forceinline__ void wait_async_le3() { asm volatile("s_wait_asynccnt 3" ::: "memory"); }
__device__ __forceinline__ void wait_async_0()  { asm volatile("s_wait_asynccnt 0" ::: "memory"); }
__device__ __forceinline__ uint32_t lds_off(const void* p) { return (uint32_t)(uintptr_t)p; }

// ---- kernel 1: compact u_t1 column 0 (2 nodes/thread, b128 NT load), zero accumulators ----
__global__ void k_init(const float* __restrict__ x_t1, float* __restrict__ u1,
                       float* __restrict__ sum, unsigned* __restrict__ cnt,
                       float* __restrict__ d2, int npair, int n) {
    int i = blockIdx.x * blockDim.x + threadIdx.x;
    if (i >= npair) return;
    int base = 2 * i;
    if (base + 1 < n) {
        vf4 x = __builtin_nontemporal_load((const vf4*)x_t1 + i);  // nodes base, base+1 (both cols)
        vf2 u; u.x = x.x; u.y = x.z;
        *(vf2*)(u1 + base) = u;
        vf2 z = {0.0f, 0.0f};
        *(vf2*)(sum + base) = z;
        *(vf2*)(d2  + base) = z;
        vu2 zi = {0u, 0u};
        *(vu2*)(cnt + base) = zi;
    } else {
        u1[base]  = __builtin_nontemporal_load(x_t1 + 2 * (size_t)base);
        sum[base] = 0.0f; cnt[base] = 0u; d2[base] = 0.0f;
    }
}

// ---- edge pass: diff=(u[dst]-u[src])/len scattered into sum[dst] (+count on pass 1) ----
template <bool ADD_COUNT>
__global__ void k_edge(const int* __restrict__ src, const int* __restrict__ dst,
                       const float* __restrict__ len, const float* __restrict__ u,
                       float* __restrict__ sum, unsigned* __restrict__ cnt,
                       int E, int ntiles, int tilesPerBlock) {
    __shared__ __align__(16) int   s_src[2][TILE];
    __shared__ __align__(16) int   s_dst[2][TILE];
    __shared__ __align__(16) float s_len[2][TILE];

    const int tid = threadIdx.x;
    const long t0 = (long)blockIdx.x * tilesPerBlock;
    long rem = (long)ntiles - t0;
    const int nt = (rem < (long)tilesPerBlock) ? (int)rem : tilesPerBlock;
    if (nt <= 0) return;

    long emax = (long)E - 2;                 // aligned clamp target for b64 copies
    emax &= ~1L;
    if (emax < 0) emax = 0;

    auto issue = [&](long tile, int b) {
        long e = tile * (long)TILE + 2 * tid;
        if (e > emax) e = emax;              // uniform issue, guarded commit below
        async_ld_b64(lds_off(&s_src[b][2 * tid]), src + e);
        async_ld_b64(lds_off(&s_dst[b][2 * tid]), dst + e);
        async_ld_b64(lds_off(&s_len[b][2 * tid]), len + e);
    };

    issue(t0, 0);
    for (int i = 0; i < nt; ++i) {
        const bool hasNext = (i + 1 < nt);
        if (hasNext) issue(t0 + i + 1, (i + 1) & 1);
        if (i + 2 < nt) {                    // warm L2 two tiles ahead
            long e2 = (t0 + i + 2) * (long)TILE + 2 * tid;
            if (e2 > emax) e2 = emax;
            __builtin_prefetch(src + e2);
            __builtin_prefetch(dst + e2);
            __builtin_prefetch(len + e2);
        }
        if (hasNext) wait_async_le3(); else wait_async_0();

        const int b = i & 1;
        const int   s0 = s_src[b][2 * tid], s1 = s_src[b][2 * tid + 1];
        const int   d0 = s_dst[b][2 * tid], d1 = s_dst[b][2 * tid + 1];
        const float l0 = s_len[b][2 * tid], l1 = s_len[b][2 * tid + 1];

        // random gathers: default (RT) policy -> L2-resident 16MB node array
        const float diff0 = (u[d0] - u[s0]) / l0;
        const float diff1 = (u[d1] - u[s1]) / l1;

        const long e = (t0 + i) * (long)TILE + 2 * tid;
        if (e < E) {
            unsafeAtomicAdd(&sum[d0], diff0);          // global_atomic_add_f32, no return
            if (ADD_COUNT) atomicAdd(&cnt[d0], 1u);
        }
        if (e + 1 < E) {
            unsafeAtomicAdd(&sum[d1], diff1);
            if (ADD_COUNT) atomicAdd(&cnt[d1], 1u);
        }
    }
}

// ---- kernel 3: in-place segment mean (2 nodes/thread) ----
__global__ void k_div(float* __restrict__ sum, const unsigned* __restrict__ cnt,
                      int npair, int n) {
    int i = blockIdx.x * blockDim.x + threadIdx.x;
    if (i >= npair) return;
    int base = 2 * i;
    if (base + 1 < n) {
        vf2 s = *(const vf2*)(sum + base);
        vu2 c = *(const vu2*)(cnt + base);
        s.x = s.x / fmaxf((float)c.x, 1.0f);
        s.y = s.y / fmaxf((float)c.y, 1.0f);
        *(vf2*)(sum + base) = s;
    } else {
        sum[base] = sum[base] / fmaxf((float)cnt[base], 1.0f);
    }
}

// ---- kernel 5: fuse temporal + advection - dissipation, mask, NT in/out ----
__global__ void k_final(const float* __restrict__ x_t, const float* __restrict__ x_t1,
                        const float* __restrict__ du, const float* __restrict__ d2sum,
                        const unsigned* __restrict__ cnt, const float* __restrict__ mask,
                        float* __restrict__ out, int npair, int n) {
    int i = blockIdx.x * blockDim.x + threadIdx.x;
    if (i >= npair) return;
    int base = 2 * i;
    if (base + 1 < n) {
        vf4 xt  = __builtin_nontemporal_load((const vf4*)x_t  + i);
        vf4 xt1 = __builtin_nontemporal_load((const vf4*)x_t1 + i);
        vf2 m   = __builtin_nontemporal_load((const vf2*)mask + i);
        vf2 dv  = *(const vf2*)(du + base);     // L2-resident: default policy
        vf2 d2  = *(const vf2*)(d2sum + base);
        vu2 c   = *(const vu2*)(cnt + base);
        float c0 = fmaxf((float)c.x, 1.0f);
        float c1 = fmaxf((float)c.y, 1.0f);
        float l0 = (xt.x - xt1.x) / DELTA_T + dv.x * xt1.x - MU * (d2.x / c0);
        float l1 = (xt.z - xt1.z) / DELTA_T + dv.y * xt1.z - MU * (d2.y / c1);
        vf2 o; o.x = l0 * m.x; o.y = l1 * m.y;
        __builtin_nontemporal_store(o, (vf2*)out + i);
    } else {
        float ut  = __builtin_nontemporal_load(x_t  + 2 * (size_t)base);
        float ut1 = __builtin_nontemporal_load(x_t1 + 2 * (size_t)base);
        float m   = __builtin_nontemporal_load(mask + base);
        float c   = fmaxf((float)cnt[base], 1.0f);
        float loss = (ut - ut1) / DELTA_T + du[base] * ut1 - MU * (d2sum[base] / c);
        __builtin_nontemporal_store(loss * m, out + base);
    }
}

extern "C" void kernel_launch(void* const* d_in, const int* in_sizes, int n_in,
                              void* d_out, int out_size, void* d_ws, size_t ws_size,
                              hipStream_t stream) {
    const float* x_t   = (const float*)d_in[0];
    const float* x_t1  = (const float*)d_in[1];
    const int*   eidx  = (const int*)d_in[2];   // (2, E) row-major: [src row | dst row]
    const float* eattr = (const float*)d_in[3];
    const float* mask  = (const float*)d_in[4];
    float* out = (float*)d_out;

    const int N = in_sizes[0] / 2;   // x_t is (N, 2)
    const int E = in_sizes[3];       // edge_attr is (E, 1)
    const int* src = eidx;
    const int* dst = eidx + (size_t)E;

    // workspace: 4 node arrays (u1c, sum->du, cnt, d2sum) = 4*N*4 bytes (64 MB)
    char* ws = (char*)d_ws;
    float*    u1  = (float*)(ws + 0 * (size_t)N * 4);
    float*    sum = (float*)(ws + 1 * (size_t)N * 4);
    unsigned* cnt = (unsigned*)(ws + 2 * (size_t)N * 4);
    float*    d2  = (float*)(ws + 3 * (size_t)N * 4);

    const int npair = (N + 1) / 2;
    const int nblk  = (npair + NTHREADS - 1) / NTHREADS;
    k_init<<<nblk, NTHREADS, 0, stream>>>(x_t1, u1, sum, cnt, d2, npair, N);

    const int ntiles = (E + TILE - 1) / TILE;
    int eblocks = 4096;
    if (eblocks > ntiles) eblocks = ntiles;
    const int tpb = (ntiles + eblocks - 1) / eblocks;

    // pass 1: du_sum + counts (counts depend only on dst -> computed once)
    k_edge<true ><<<eblocks, NTHREADS, 0, stream>>>(src, dst, eattr, u1, sum, cnt, E, ntiles, tpb);
    k_div<<<nblk, NTHREADS, 0, stream>>>(sum, cnt, npair, N);   // sum becomes du
    // pass 2: d2u_sum over du
    k_edge<false><<<eblocks, NTHREADS, 0, stream>>>(src, dst, eattr, sum, d2, cnt, E, ntiles, tpb);

    k_final<<<nblk, NTHREADS, 0, stream>>>(x_t, x_t1, sum, d2, cnt, mask, out, npair, N);
}